// DiscriminativeLoss_43344809951808
// MI455X (gfx1250) — compile-verified
//
#include <hip/hip_runtime.h>

typedef float v2f __attribute__((ext_vector_type(2)));
typedef float v8f __attribute__((ext_vector_type(8)));

#define C_CLUSTERS 32
#define DIM 16

// workspace layout (floats)
#define WS_SUMS   0      // 32*16
#define WS_CNT    512    // 32
#define WS_MEANS  544    // 32*16
#define WS_W      1056   // 32
#define WS_VAR    1088
#define WS_DIST   1089
#define WS_REG    1090
#define WS_TOTAL  1096

// ---------------------------------------------------------------- zero scratch
__global__ __launch_bounds__(256) void k_zero(float* __restrict__ ws) {
    for (int i = threadIdx.x; i < WS_TOTAL; i += 256) ws[i] = 0.0f;
}

// ---------------------------------------------------------------- pass 1
// Segment sums via one-hot WMMA:  D(16 clusters x 16 dims) += A(16x4 onehot) * B(4 pts x 16 dims)
// Two cluster groups (0-15, 16-31) -> 2 WMMAs per K-step of 4 points.
// Labels are wave-uniform scalar loads; B is a per-lane transposed f32 load.
// Streams features front-to-back, so the high-index tail of the array is what
// remains resident in the 192 MB L2 for pass 3 to reuse.
__global__ __launch_bounds__(256) void k_seg_sums(const float* __restrict__ f,
                                                  const int* __restrict__ lb,
                                                  float* __restrict__ ws, int n) {
    const int lane = threadIdx.x & 31;
    const int wave = __builtin_amdgcn_readfirstlane((blockIdx.x * blockDim.x + threadIdx.x) >> 5);
    const int nwaves = (gridDim.x * blockDim.x) >> 5;
    const int h = lane >> 4;   // half-wave select
    const int m = lane & 15;   // cluster-in-group (A) / dim column (B)

    v8f c0 = {0.f, 0.f, 0.f, 0.f, 0.f, 0.f, 0.f, 0.f};
    v8f c1 = {0.f, 0.f, 0.f, 0.f, 0.f, 0.f, 0.f, 0.f};
    float cntf = 0.0f;

    const int laneOff = h * 2 * DIM + m;   // B: vgpr j holds point (k*4 + 2h + j), dim m
    // n is a multiple of 32 (4M); each wave iteration consumes 32 points.
    for (int it = wave; it * 32 < n; it += nwaves) {
        const int base = it * 32;
        const float* fp = f + (size_t)base * DIM + laneOff;
        const int*   lp = lb + base;   // uniform -> s_load
#pragma unroll
        for (int k = 0; k < 8; ++k) {
            const int l0 = lp[k * 4 + 0];
            const int l1 = lp[k * 4 + 1];
            const int l2 = lp[k * 4 + 2];
            const int l3 = lp[k * 4 + 3];

            v2f b;
            b.x = fp[(k * 4 + 0) * DIM];
            b.y = fp[(k * 4 + 1) * DIM];

            const int la = h ? l2 : l0;   // label of point k*4 + 2h + 0
            const int lc = h ? l3 : l1;   // label of point k*4 + 2h + 1
            v2f a0, a1;
            a0.x = (la == m)      ? 1.0f : 0.0f;
            a0.y = (lc == m)      ? 1.0f : 0.0f;
            a1.x = (la == m + 16) ? 1.0f : 0.0f;
            a1.y = (lc == m + 16) ? 1.0f : 0.0f;

            c0 = __builtin_amdgcn_wmma_f32_16x16x4_f32(false, a0, false, b, (short)0, c0, false, false);
            c1 = __builtin_amdgcn_wmma_f32_16x16x4_f32(false, a1, false, b, (short)0, c1, false, false);

            cntf += (lane == l0 ? 1.0f : 0.0f) + (lane == l1 ? 1.0f : 0.0f)
                  + (lane == l2 ? 1.0f : 0.0f) + (lane == l3 ? 1.0f : 0.0f);
        }
    }
    // D layout: vgpr r -> cluster r + 8h (+16 for group 1), dim = m
#pragma unroll
    for (int r = 0; r < 8; ++r) {
        atomicAdd(&ws[WS_SUMS + (r + 8 * h) * DIM + m],       c0[r]);
        atomicAdd(&ws[WS_SUMS + (r + 8 * h + 16) * DIM + m],  c1[r]);
    }
    atomicAdd(&ws[WS_CNT + lane], cntf);
}

// ---------------------------------------------------------------- pass 2
// Single full wave: means, weights, reg loss, and inter-cluster loss via
// WMMA Gram matrix G = M * M^T (32x32 from four 16x16 f32 WMMA tiles).
__global__ __launch_bounds__(32) void k_means_dist(float* __restrict__ ws) {
    __shared__ float sM[C_CLUSTERS * DIM];
    __shared__ float sN2[C_CLUSTERS];

    const int lane = threadIdx.x;          // 0..31 = cluster id
    const float cnt  = ws[WS_CNT + lane];
    const float safe = fmaxf(cnt, 1.0f);
    float n2 = 0.0f;
#pragma unroll
    for (int d = 0; d < DIM; ++d) {
        const float mu = ws[WS_SUMS + lane * DIM + d] / safe;
        sM[lane * DIM + d] = mu;
        ws[WS_MEANS + lane * DIM + d] = mu;
        n2 += mu * mu;
    }
    sN2[lane] = n2;
    ws[WS_W + lane] = 1.0f / (32.0f * safe);

    // reg loss = mean centroid norm
    float reg = sqrtf(n2);
#pragma unroll
    for (int off = 16; off > 0; off >>= 1) reg += __shfl_down(reg, off, 32);
    if (lane == 0) ws[WS_REG] = reg / 32.0f;
    __syncthreads();

    const int h = lane >> 4, m = lane & 15;
    float acc = 0.0f;
#pragma unroll
    for (int gi = 0; gi < 2; ++gi) {
#pragma unroll
        for (int gj = 0; gj < 2; ++gj) {
            v8f c = {0.f, 0.f, 0.f, 0.f, 0.f, 0.f, 0.f, 0.f};
#pragma unroll
            for (int k = 0; k < 4; ++k) {
                v2f a, b;
                a.x = sM[(gi * 16 + m) * DIM + k * 4 + 2 * h + 0];
                a.y = sM[(gi * 16 + m) * DIM + k * 4 + 2 * h + 1];
                b.x = sM[(gj * 16 + m) * DIM + k * 4 + 2 * h + 0];
                b.y = sM[(gj * 16 + m) * DIM + k * 4 + 2 * h + 1];
                c = __builtin_amdgcn_wmma_f32_16x16x4_f32(false, a, false, b, (short)0, c, false, false);
            }
            const int   cj  = gj * 16 + m;
            const float n2j = sN2[cj];
#pragma unroll
            for (int r = 0; r < 8; ++r) {
                const int   ci  = gi * 16 + r + 8 * h;
                const float d2  = sN2[ci] + n2j - 2.0f * c[r];
                const float pd  = sqrtf(fmaxf(d2, 0.0f) + 1e-12f);
                const float hh  = fmaxf(3.0f - pd, 0.0f);   // 2*DELTA_DIST = 3.0
                acc += (ci == cj) ? 0.0f : hh * hh;
            }
        }
    }
#pragma unroll
    for (int off = 16; off > 0; off >>= 1) acc += __shfl_down(acc, off, 32);
    if (lane == 0) ws[WS_DIST] = acc / 992.0f;   // C*(C-1)
}

// ---------------------------------------------------------------- pass 3
// var_loss streaming pass: one point per lane, means + weights staged in LDS.
// Iterates BACK-TO-FRONT: pass 1 finished streaming the high-index end of the
// 256 MB feature array, so that tail is still resident in the 192 MB L2 —
// consuming it first (before this pass's own stream evicts it) saves up to
// ~190 MB of HBM traffic on a purely bandwidth-bound pass.
__global__ __launch_bounds__(256) void k_var(const float* __restrict__ f,
                                             const int* __restrict__ lb,
                                             float* __restrict__ ws, int n) {
    __shared__ float sM[C_CLUSTERS * DIM];
    __shared__ float sW[C_CLUSTERS];
    for (int i = threadIdx.x; i < C_CLUSTERS * DIM; i += 256) sM[i] = ws[WS_MEANS + i];
    if (threadIdx.x < C_CLUSTERS) sW[threadIdx.x] = ws[WS_W + threadIdx.x];
    __syncthreads();

    float acc = 0.0f;
    const int stride = gridDim.x * blockDim.x;
    for (int q = blockIdx.x * blockDim.x + threadIdx.x; q < n; q += stride) {
        const int p = n - 1 - q;   // reverse order for L2 tail reuse
        const float4* fp = (const float4*)(f + (size_t)p * DIM);
        const float4  f0 = fp[0], f1 = fp[1], f2 = fp[2], f3 = fp[3];
        const int     l  = lb[p];
        const float4* mp = (const float4*)&sM[l * DIM];
        const float4  m0 = mp[0], m1 = mp[1], m2 = mp[2], m3 = mp[3];
        float d2 = 0.0f, t;
        t = f0.x - m0.x; d2 += t * t;  t = f0.y - m0.y; d2 += t * t;
        t = f0.z - m0.z; d2 += t * t;  t = f0.w - m0.w; d2 += t * t;
        t = f1.x - m1.x; d2 += t * t;  t = f1.y - m1.y; d2 += t * t;
        t = f1.z - m1.z; d2 += t * t;  t = f1.w - m1.w; d2 += t * t;
        t = f2.x - m2.x; d2 += t * t;  t = f2.y - m2.y; d2 += t * t;
        t = f2.z - m2.z; d2 += t * t;  t = f2.w - m2.w; d2 += t * t;
        t = f3.x - m3.x; d2 += t * t;  t = f3.y - m3.y; d2 += t * t;
        t = f3.z - m3.z; d2 += t * t;  t = f3.w - m3.w; d2 += t * t;
        const float hinge = fmaxf(sqrtf(d2) - 0.5f, 0.0f);   // DELTA_VAR = 0.5
        acc += hinge * hinge * sW[l];
    }
#pragma unroll
    for (int off = 16; off > 0; off >>= 1) acc += __shfl_down(acc, off, 32);
    if ((threadIdx.x & 31) == 0) atomicAdd(&ws[WS_VAR], acc);
}

// ---------------------------------------------------------------- final combine
__global__ void k_final(const float* __restrict__ ws, float* __restrict__ out) {
    const float var  = ws[WS_VAR];
    const float dist = ws[WS_DIST];
    const float reg  = ws[WS_REG];
    out[0] = var + dist + 0.001f * reg;  // ALPHA=BETA=1, GAMMA=0.001
    out[1] = var;
    out[2] = dist;
    out[3] = reg;
}

extern "C" void kernel_launch(void* const* d_in, const int* in_sizes, int n_in,
                              void* d_out, int out_size, void* d_ws, size_t ws_size,
                              hipStream_t stream) {
    const float* f  = (const float*)d_in[0];
    const int*   lb = (const int*)d_in[1];
    float*       ws = (float*)d_ws;
    float*       out = (float*)d_out;
    const int n = in_sizes[1];   // number of points (labels count)

    k_zero<<<1, 256, 0, stream>>>(ws);
    k_seg_sums<<<960, 256, 0, stream>>>(f, lb, ws, n);
    k_means_dist<<<1, 32, 0, stream>>>(ws);
    k_var<<<960, 256, 0, stream>>>(f, lb, ws, n);
    k_final<<<1, 1, 0, stream>>>(ws, out);
}